// MixDimensionEmbeddingBag_13194139533840
// MI455X (gfx1250) — compile-verified
//
#include <hip/hip_runtime.h>

// ---------------------------------------------------------------------------
// MixDimensionEmbeddingBag for MI455X (gfx1250, wave32)
//   out[b,:] = sum_f t0[x[b,f]+f*V]                      (8 fields, d=64)
//            + (sum_f t1[x[b,8+f]+f*V]) @ W1^T + 8*b1    (8 fields, d=32)
//            + (sum_f t2[x[b,16+f]+f*V]) @ W2^T + 10*b2  (10 fields, d=16)
// Projections done ONCE per row on the field-summed vector (linearity),
// executed with V_WMMA_F32_16X16X4_F32.
// ---------------------------------------------------------------------------

typedef __attribute__((ext_vector_type(2))) float v2f;
typedef __attribute__((ext_vector_type(8))) float v8f;

#define BATCH      16384
#define NFIELDS    26
#define VOCAB      100000
#define DIMOUT     64
#define ROWS_PER_WG 16

// LDS row strides in floats (padded to spread banks; multiples of 4 keep
// float4/float2 alignment for b128/b64 LDS ops).
#define ACC_STRIDE 68   // 16 x 64 block-0 accumulator
#define S1_STRIDE  36   // 16 x 32 block-1 field-sum
#define S2_STRIDE  20   // 16 x 16 block-2 field-sum

static __device__ __forceinline__ void add4(float4& a, const float4 b) {
    a.x += b.x; a.y += b.y; a.z += b.z; a.w += b.w;
}

__global__ __launch_bounds__(64)
void mixdim_embbag_kernel(const int*   __restrict__ x,
                          const float* __restrict__ t0,
                          const float* __restrict__ t1,
                          const float* __restrict__ t2,
                          const float* __restrict__ W1,   // [64][32] row-major
                          const float* __restrict__ b1,   // [64]
                          const float* __restrict__ W2,   // [64][16] row-major
                          const float* __restrict__ b2,   // [64]
                          float*       __restrict__ out)  // [B][64]
{
    __shared__ float acc_lds[ROWS_PER_WG * ACC_STRIDE];
    __shared__ float s1_lds [ROWS_PER_WG * S1_STRIDE];
    __shared__ float s2_lds [ROWS_PER_WG * S2_STRIDE];

    const int tid      = threadIdx.x;       // 0..63  (2 waves of 32)
    const int row      = tid >> 2;          // 0..15  (one batch row per 4 lanes)
    const int q        = tid & 3;           // quarter of the row's columns
    const int row_base = blockIdx.x * ROWS_PER_WG;
    const long xrow    = (long)(row_base + row) * NFIELDS;

    // ---------------- Phase 1: gather + field-sum (all 64 lanes) -----------
    // Block 0: dim 64, 8 fields. Lane covers 16 consecutive cols of one row.
    {
        const int colbase = q * 16;
        float4 a0 = {0.f,0.f,0.f,0.f}, a1 = a0, a2 = a0, a3 = a0;
        #pragma unroll
        for (int f = 0; f < 8; ++f) {
            const long idx = (long)x[xrow + f] + (long)f * VOCAB;
            const float4* src = (const float4*)(t0 + idx * 64 + colbase);
            add4(a0, src[0]); add4(a1, src[1]); add4(a2, src[2]); add4(a3, src[3]);
        }
        float4* dst = (float4*)(acc_lds + row * ACC_STRIDE + colbase);
        dst[0] = a0; dst[1] = a1; dst[2] = a2; dst[3] = a3;
    }

    // Block 1: dim 32, 8 fields. Lane covers 8 cols of one row.
    {
        const int colbase = q * 8;
        float4 a0 = {0.f,0.f,0.f,0.f}, a1 = a0;
        #pragma unroll
        for (int f = 0; f < 8; ++f) {
            const long idx = (long)x[xrow + 8 + f] + (long)f * VOCAB;
            const float4* src = (const float4*)(t1 + idx * 32 + colbase);
            add4(a0, src[0]); add4(a1, src[1]);
        }
        float4* dst = (float4*)(s1_lds + row * S1_STRIDE + colbase);
        dst[0] = a0; dst[1] = a1;
    }

    // Block 2: dim 16, 10 fields. Lane covers 4 cols of one row.
    {
        const int colbase = q * 4;
        float4 a0 = {0.f,0.f,0.f,0.f};
        #pragma unroll
        for (int f = 0; f < 10; ++f) {
            const long idx = (long)x[xrow + 16 + f] + (long)f * VOCAB;
            add4(a0, *(const float4*)(t2 + idx * 16 + colbase));
        }
        *(float4*)(s2_lds + row * S2_STRIDE + colbase) = a0;
    }

    __syncthreads();

    // ---------------- Phase 2: WMMA projections (per wave, EXEC all-1s) ----
    // Each of the 2 waves owns 2 of the 4 16-wide output-column tiles.
    // D = A(16xK) x B(Kx16) + C with V_WMMA_F32_16X16X4_F32 chained over K.
    //   A lane layout: M = lane&15, K-pair = 2*(lane>>4)   (ISA 7.12.2)
    //   B lane layout: N = lane&15, K-pair = 2*(lane>>4)
    //   C/D layout:    N = lane&15, M = vgpr + 8*(lane>>4)
    const int ln = tid & 31;
    const int wv = tid >> 5;
    const int hi = ln >> 4;          // 0/1: which K-pair / which M half
    const int nn = ln & 15;          // output column within tile
    const int mm = ln & 15;          // A-matrix row

    #pragma unroll
    for (int t = 0; t < 2; ++t) {
        const int n0  = wv * 32 + t * 16;
        const int col = n0 + nn;

        // Seed accumulator with the identity-block sums (C/D layout).
        v8f c;
        #pragma unroll
        for (int j = 0; j < 8; ++j)
            c[j] = acc_lds[(j + 8 * hi) * ACC_STRIDE + col];

        // Block-1 projection: out += s1 @ W1^T   (K = 32 -> 8 WMMAs)
        #pragma unroll
        for (int k0 = 0; k0 < 32; k0 += 4) {
            v2f a = *(const v2f*)(s1_lds + mm * S1_STRIDE + k0 + 2 * hi);
            v2f b = *(const v2f*)(W1 + col * 32 + k0 + 2 * hi);
            c = __builtin_amdgcn_wmma_f32_16x16x4_f32(
                    false, a, false, b, (short)0, c, false, false);
        }

        // Block-2 projection: out += s2 @ W2^T   (K = 16 -> 4 WMMAs)
        #pragma unroll
        for (int k0 = 0; k0 < 16; k0 += 4) {
            v2f a = *(const v2f*)(s2_lds + mm * S2_STRIDE + k0 + 2 * hi);
            v2f b = *(const v2f*)(W2 + col * 16 + k0 + 2 * hi);
            c = __builtin_amdgcn_wmma_f32_16x16x4_f32(
                    false, a, false, b, (short)0, c, false, false);
        }

        // Fold in biases (8 copies of b1, 10 of b2 per reference semantics)
        // and write out in the C/D register layout.
        const float bias = 8.0f * b1[col] + 10.0f * b2[col];
        #pragma unroll
        for (int j = 0; j < 8; ++j)
            out[(long)(row_base + j + 8 * hi) * DIMOUT + col] = c[j] + bias;
    }
}

extern "C" void kernel_launch(void* const* d_in, const int* in_sizes, int n_in,
                              void* d_out, int out_size, void* d_ws, size_t ws_size,
                              hipStream_t stream) {
    (void)in_sizes; (void)n_in; (void)out_size; (void)d_ws; (void)ws_size;
    const int*   x  = (const int*)  d_in[0];
    const float* t0 = (const float*)d_in[1];
    const float* t1 = (const float*)d_in[2];
    const float* t2 = (const float*)d_in[3];
    const float* W1 = (const float*)d_in[4];
    const float* b1 = (const float*)d_in[5];
    const float* W2 = (const float*)d_in[6];
    const float* b2 = (const float*)d_in[7];
    float* out = (float*)d_out;

    dim3 grid(BATCH / ROWS_PER_WG);   // 1024 workgroups
    dim3 block(64);                   // 2 wave32s per workgroup
    mixdim_embbag_kernel<<<grid, block, 0, stream>>>(x, t0, t1, t2, W1, b1, W2, b2, out);
}